// EntitiesAsExperts_60421599920498
// MI455X (gfx1250) — compile-verified
//
#include <hip/hip_runtime.h>

// ---------------------------------------------------------------------------
// Problem constants (from reference)
// ---------------------------------------------------------------------------
#define BB      8
#define SS      512
#define DEMB    768
#define DENT    256
#define NENT    500000
#define MM      256          // mentions
#define MAXK    128          // k=100 in reference; sized up

// scores kernel tiling
#define CHUNK   4096                         // entities per block
#define NCHUNK  ((NENT + CHUNK - 1) / CHUNK) // 123
#define MTILE   64                           // mentions per block (4 wmma tiles)
#define NSUPER  (MM / MTILE)                 // 4
#define BBINS   256                          // per-chunk histogram bins
#define CAP     192                          // candidate capacity per (mention,chunk)

// merge kernel
#define MBINS   8192
#define MCAP    1024

typedef __attribute__((ext_vector_type(2))) float v2f;
typedef __attribute__((ext_vector_type(8))) float v8f;

__device__ __forceinline__ int score_bin(float v, int bins) {
  float x = (v + 4.0f) * (bins * 0.125f);   // bins / 8 over [-4, 4]
  int b = (int)x;
  b = b < 0 ? 0 : b;
  b = b >= bins ? bins - 1 : b;
  return b;
}

// ---------------------------------------------------------------------------
// Kernel Z: zero the output tensor
// ---------------------------------------------------------------------------
__global__ void zero_kernel(float* __restrict__ y, int n) {
  int i = blockIdx.x * blockDim.x + threadIdx.x;
  if (i < n) y[i] = 0.0f;
}

// ---------------------------------------------------------------------------
// Kernel A: gather span endpoints and compute pseudo = span @ Wf_w^T + Wf_b
// One block per mention, 256 threads (one per d_ent output).
// ---------------------------------------------------------------------------
__global__ void pseudo_kernel(const float* __restrict__ X,
                              const int* __restrict__ mb,
                              const int* __restrict__ mbeg,
                              const int* __restrict__ mend,
                              const float* __restrict__ Wf_w,
                              const float* __restrict__ Wf_b,
                              float* __restrict__ pseudo) {
  __shared__ float span[2 * DEMB];
  const int m = blockIdx.x;
  const int tid = threadIdx.x;
  const int b = mb[m], s1 = mbeg[m], s2 = mend[m];
  const float* x1 = X + ((size_t)b * SS + s1) * DEMB;
  const float* x2 = X + ((size_t)b * SS + s2) * DEMB;
  for (int i = tid; i < DEMB; i += 256) {
    span[i] = x1[i];
    span[DEMB + i] = x2[i];
  }
  __syncthreads();
  const float* wr = Wf_w + (size_t)tid * (2 * DEMB);
  float acc = Wf_b[tid];
  for (int i = 0; i < 2 * DEMB; ++i) acc = fmaf(wr[i], span[i], acc);
  pseudo[m * DENT + tid] = acc;
}

// ---------------------------------------------------------------------------
// Kernel B: scores = pseudo @ E_w via V_WMMA_F32_16X16X4_F32, fused chunk-local
// top-(>=k) selection.
//   grid = (NCHUNK, NSUPER), block = 256 threads = 8 waves.
//   Each wave owns TWO 16-entity column strips (2 WMMAs per A-fragment fetch);
//   block covers 64 mentions x 4096 entities.  Phase 0 histograms all scores
//   (64 x 256 bins in LDS), a per-mention chunk-local threshold bin is
//   derived, phase 1 recomputes the identical GEMM and appends candidates
//   >= threshold to global workspace (chunk-local top-100 is a superset of
//   the global top-100 restricted to the chunk => exact merge possible).
// LDS: pseudo tile 64x260 f32 (padded stride: bank-conflict-free b64 A-frag
// reads) + 64x256 u32 histogram + counters ~= 133 KB -> 2 blocks per 320 KB
// WGP (4 waves/SIMD for latency hiding).
// ---------------------------------------------------------------------------
#define PSA_STRIDE 260
#define SMEM_B ((MTILE * PSA_STRIDE + MTILE * BBINS + MTILE + MTILE) * 4)

__global__ __launch_bounds__(256) void score_topk_kernel(
    const float* __restrict__ pseudo, const float* __restrict__ Ew,
    const int* __restrict__ kptr, float* __restrict__ cand_val,
    int* __restrict__ cand_idx, int* __restrict__ cand_cnt) {
  extern __shared__ char smem[];
  float* psA = (float*)smem;                                   // [MTILE][260]
  unsigned* hist = (unsigned*)(smem + MTILE * PSA_STRIDE * 4); // [MTILE][BBINS]
  unsigned* ccnt = hist + MTILE * BBINS;                       // [MTILE]
  int* thrb = (int*)(ccnt + MTILE);                            // [MTILE]

  const int chunk = blockIdx.x;
  const int mbase = blockIdx.y * MTILE;
  const int tid = threadIdx.x;
  const int lane = tid & 31;
  const int wave = tid >> 5;
  const int hsel = lane >> 4;     // which half-wave (K pair select)
  const int l15 = lane & 15;      // row/col within fragment
  const int K = *kptr;

  // Load pseudo tile and clear LDS state.
  for (int i = tid; i < MTILE * DENT; i += 256) {
    int m = i >> 8, kk = i & 255;
    psA[m * PSA_STRIDE + kk] = pseudo[(mbase + m) * DENT + kk];
  }
  for (int i = tid; i < MTILE * BBINS; i += 256) hist[i] = 0u;
  if (tid < MTILE) { ccnt[tid] = 0u; thrb[tid] = 0; }
  __syncthreads();

  const int nbase = chunk * CHUNK;

  for (int phase = 0; phase < 2; ++phase) {
    for (int pass = 0; pass < CHUNK / 256; ++pass) {     // 16 passes
      const int n0 = nbase + pass * 256 + wave * 32;
      const int ncol0 = n0 + l15;
      const int ncol1 = n0 + 16 + l15;
      const bool valid0 = ncol0 < NENT;
      const bool valid1 = ncol1 < NENT;
      const int nc0 = valid0 ? ncol0 : 0;   // clamp: garbage lanes dropped later
      const int nc1 = valid1 ? ncol1 : 0;

      v8f acc[4][2];
#pragma unroll
      for (int t = 0; t < 4; ++t) {
        acc[t][0] = (v8f)(0.0f);
        acc[t][1] = (v8f)(0.0f);
      }

      // K loop: 8 x V_WMMA_F32_16X16X4_F32 per step; each A fragment feeds
      // two column strips, both B fragments shared by 4 mention tiles.
      for (int k = 0; k < DENT; k += 4) {
        const size_t r0 = (size_t)(k + 2 * hsel) * NENT;
        v2f bf0, bf1;
        bf0.x = Ew[r0 + nc0];
        bf0.y = Ew[r0 + NENT + nc0];
        bf1.x = Ew[r0 + nc1];
        bf1.y = Ew[r0 + NENT + nc1];
#pragma unroll
        for (int t = 0; t < 4; ++t) {
          const float* ap = &psA[(t * 16 + l15) * PSA_STRIDE + k + 2 * hsel];
          v2f af;
          af.x = ap[0];
          af.y = ap[1];
          acc[t][0] = __builtin_amdgcn_wmma_f32_16x16x4_f32(
              false, af, false, bf0, (short)0, acc[t][0], false, false);
          acc[t][1] = __builtin_amdgcn_wmma_f32_16x16x4_f32(
              false, af, false, bf1, (short)0, acc[t][1], false, false);
        }
      }

      // Consume the 64 scores this lane holds.
#pragma unroll
      for (int c = 0; c < 2; ++c) {
        const bool valid = c ? valid1 : valid0;
        const int ncol = c ? ncol1 : ncol0;
#pragma unroll
        for (int t = 0; t < 4; ++t) {
#pragma unroll
          for (int i = 0; i < 8; ++i) {
            const float s = acc[t][c][i];
            const int mloc = t * 16 + i + 8 * hsel; // C layout: VGPR i -> M=i/i+8
            if (valid) {
              const int bin = score_bin(s, BBINS);
              if (phase == 0) {
                atomicAdd(&hist[mloc * BBINS + bin], 1u);
              } else if (bin >= thrb[mloc]) {
                const unsigned slot = atomicAdd(&ccnt[mloc], 1u);
                if (slot < CAP) {
                  const size_t o =
                      ((size_t)(mbase + mloc) * NCHUNK + chunk) * CAP + slot;
                  cand_val[o] = s;
                  cand_idx[o] = ncol;
                }
              }
            }
          }
        }
      }
    }

    if (phase == 0) {
      __syncthreads();
      if (tid < MTILE) {   // per-mention threshold: highest bin with cum >= K
        unsigned cum = 0;
        int res = 0;
        for (int b = BBINS - 1; b >= 0; --b) {
          cum += hist[tid * BBINS + b];
          if (cum >= (unsigned)K) { res = b; break; }
        }
        thrb[tid] = (cum >= (unsigned)K) ? res : 0;
      }
      __syncthreads();
    }
  }

  __syncthreads();
  if (tid < MTILE) {
    unsigned c = ccnt[tid];
    if (c > CAP) c = CAP;
    cand_cnt[(size_t)(mbase + tid) * NCHUNK + chunk] = (int)c;
  }
}

// ---------------------------------------------------------------------------
// Kernel C: exact per-mention merge of chunk candidates -> top-k + softmax.
// One block (256 threads) per mention.
// ---------------------------------------------------------------------------
__global__ __launch_bounds__(256) void merge_topk_kernel(
    const float* __restrict__ cand_val, const int* __restrict__ cand_idx,
    const int* __restrict__ cand_cnt, const int* __restrict__ kptr,
    int* __restrict__ top_idx, float* __restrict__ alpha) {
  __shared__ unsigned hist[MBINS];
  __shared__ float sval[MCAP];
  __shared__ int sidx[MCAP];
  __shared__ int scnt[NCHUNK];
  __shared__ unsigned csel;
  __shared__ int bstar;
  __shared__ float smax, ssum;

  const int m = blockIdx.x;
  const int tid = threadIdx.x;
  int K = *kptr;
  K = K < 1 ? 1 : (K > MAXK ? MAXK : K);

  for (int i = tid; i < MBINS; i += 256) hist[i] = 0u;
  for (int i = tid; i < NCHUNK; i += 256)
    scnt[i] = cand_cnt[(size_t)m * NCHUNK + i];
  if (tid == 0) csel = 0u;
  __syncthreads();

  // Pass 1: fine histogram of all candidate values.
  for (int i = tid; i < NCHUNK * CAP; i += 256) {
    const int c = i / CAP, s = i - c * CAP;
    if (s < scnt[c]) {
      const float v = cand_val[((size_t)m * NCHUNK + c) * CAP + s];
      atomicAdd(&hist[score_bin(v, MBINS)], 1u);
    }
  }
  __syncthreads();
  if (tid == 0) {
    unsigned cum = 0;
    int res = 0;
    for (int b = MBINS - 1; b >= 0; --b) {
      cum += hist[b];
      if (cum >= (unsigned)K) { res = b; break; }
    }
    bstar = (cum >= (unsigned)K) ? res : 0;
  }
  __syncthreads();

  // Pass 2: collect values at/above the threshold bin.
  for (int i = tid; i < NCHUNK * CAP; i += 256) {
    const int c = i / CAP, s = i - c * CAP;
    if (s < scnt[c]) {
      const size_t o = ((size_t)m * NCHUNK + c) * CAP + s;
      const float v = cand_val[o];
      if (score_bin(v, MBINS) >= bstar) {
        const unsigned slot = atomicAdd(&csel, 1u);
        if (slot < MCAP) {
          sval[slot] = v;
          sidx[slot] = cand_idx[o];
        }
      }
    }
  }
  __syncthreads();
  const unsigned total = csel < MCAP ? csel : MCAP;
  for (int i = (int)total + tid; i < MCAP; i += 256) {
    sval[i] = -__builtin_huge_valf();
    sidx[i] = 0x7fffffff;
  }
  __syncthreads();

  // Bitonic sort, descending by value, ties broken by ascending index
  // (matches jax.lax.top_k tie-breaking).
  for (int kk = 2; kk <= MCAP; kk <<= 1) {
    for (int j = kk >> 1; j > 0; j >>= 1) {
      for (int i = tid; i < MCAP; i += 256) {
        const int ixj = i ^ j;
        if (ixj > i) {
          const float va = sval[i], vb = sval[ixj];
          const int ia = sidx[i], ib = sidx[ixj];
          // rank_before(a, b): a sorts before b in descending order
          const bool b_before_a = (vb > va) || (vb == va && ib < ia);
          const bool a_before_b = (va > vb) || (va == vb && ia < ib);
          const bool doswap = ((i & kk) == 0) ? b_before_a : a_before_b;
          if (doswap) {
            sval[i] = vb; sval[ixj] = va;
            sidx[i] = ib; sidx[ixj] = ia;
          }
        }
      }
      __syncthreads();
    }
  }

  // Softmax over the top-K values.
  if (tid == 0) {
    const float mx = sval[0];
    float sum = 0.0f;
    for (int j = 0; j < K; ++j) sum += __expf(sval[j] - mx);
    smax = mx;
    ssum = sum;
  }
  __syncthreads();
  if (tid < K) {
    top_idx[m * MAXK + tid] = sidx[tid];
    alpha[m * MAXK + tid] = __expf(sval[tid] - smax) / ssum;
  }
}

// ---------------------------------------------------------------------------
// Kernel D: picked = sum_k alpha * E_w[:, idx]; upd = picked @ Wb_w^T + Wb_b;
// scatter to y[b, begin]. Duplicate (b,begin): highest mention index wins
// (deterministic, race-free).
// ---------------------------------------------------------------------------
__global__ __launch_bounds__(256) void decode_kernel(
    const int* __restrict__ mb, const int* __restrict__ mbeg,
    const float* __restrict__ Ew, const float* __restrict__ Wb_w,
    const float* __restrict__ Wb_b, const int* __restrict__ top_idx,
    const float* __restrict__ alpha, const int* __restrict__ kptr,
    float* __restrict__ y) {
  __shared__ float picked[DENT];
  __shared__ int winner;
  const int m = blockIdx.x;
  const int tid = threadIdx.x;
  int K = *kptr;
  K = K < 1 ? 1 : (K > MAXK ? MAXK : K);
  const int b = mb[m], beg = mbeg[m];

  if (tid == 0) winner = 1;
  __syncthreads();
  for (int mp = m + 1 + tid; mp < MM; mp += 256)
    if (mb[mp] == b && mbeg[mp] == beg) winner = 0;  // benign race, all write 0

  float acc = 0.0f;
  for (int j = 0; j < K; ++j)
    acc = fmaf(alpha[m * MAXK + j],
               Ew[(size_t)tid * NENT + top_idx[m * MAXK + j]], acc);
  picked[tid] = acc;
  __syncthreads();
  if (!winner) return;

  for (int e = tid; e < DEMB; e += 256) {
    const float* wr = Wb_w + (size_t)e * DENT;
    float u = Wb_b[e];
    for (int d = 0; d < DENT; ++d) u = fmaf(wr[d], picked[d], u);
    y[((size_t)b * SS + beg) * DEMB + e] = u;
  }
}

// ---------------------------------------------------------------------------
// Host launcher
// ---------------------------------------------------------------------------
extern "C" void kernel_launch(void* const* d_in, const int* in_sizes, int n_in,
                              void* d_out, int out_size, void* d_ws,
                              size_t ws_size, hipStream_t stream) {
  (void)in_sizes; (void)n_in; (void)out_size; (void)ws_size;
  const float* X     = (const float*)d_in[0];
  const int*   mb    = (const int*)d_in[1];
  const int*   mbeg  = (const int*)d_in[2];
  const int*   mend  = (const int*)d_in[3];
  const float* Wf_w  = (const float*)d_in[4];
  const float* Wf_b  = (const float*)d_in[5];
  const float* Wb_w  = (const float*)d_in[6];
  const float* Wb_b  = (const float*)d_in[7];
  const float* Ew    = (const float*)d_in[8];
  const int*   kptr  = (const int*)d_in[9];
  float* y = (float*)d_out;

  char* ws = (char*)d_ws;
  size_t off = 0;
  auto carve = [&](size_t bytes) {
    void* p = ws + off;
    off += (bytes + 255) & ~(size_t)255;
    return p;
  };
  float* pseudo   = (float*)carve((size_t)MM * DENT * 4);
  float* cand_val = (float*)carve((size_t)MM * NCHUNK * CAP * 4);
  int*   cand_idx = (int*)carve((size_t)MM * NCHUNK * CAP * 4);
  int*   cand_cnt = (int*)carve((size_t)MM * NCHUNK * 4);
  int*   top_idx  = (int*)carve((size_t)MM * MAXK * 4);
  float* alpha    = (float*)carve((size_t)MM * MAXK * 4);

  const int yN = BB * SS * DEMB;
  zero_kernel<<<(yN + 255) / 256, 256, 0, stream>>>(y, yN);

  pseudo_kernel<<<MM, 256, 0, stream>>>(X, mb, mbeg, mend, Wf_w, Wf_b, pseudo);

  dim3 gridB(NCHUNK, NSUPER);
  score_topk_kernel<<<gridB, 256, SMEM_B, stream>>>(pseudo, Ew, kptr, cand_val,
                                                    cand_idx, cand_cnt);

  merge_topk_kernel<<<MM, 256, 0, stream>>>(cand_val, cand_idx, cand_cnt, kptr,
                                            top_idx, alpha);

  decode_kernel<<<MM, 256, 0, stream>>>(mb, mbeg, Ew, Wb_w, Wb_b, top_idx,
                                        alpha, kptr, y);
}